// Quantizer_24653112279399
// MI455X (gfx1250) — compile-verified
//
#include <hip/hip_runtime.h>
#include <hip/hip_bf16.h>

typedef __attribute__((ext_vector_type(2))) float v2f;
typedef __attribute__((ext_vector_type(8))) float v8f;
typedef __attribute__((ext_vector_type(4))) unsigned int u32x4;
typedef __attribute__((ext_vector_type(4))) int i32x4;
typedef __attribute__((ext_vector_type(8))) int i32x8;

constexpr int BS = 16384, TPD = 4, D = 128;
constexpr int G = 4, M = 4096;
constexpr int NTILE = 128;        // x rows per block
constexpr int MTILE = 32;         // embedding rows per LDS chunk (double buffered)
constexpr int EPAD  = D + 4;      // padded LDS row stride (TDM pad reproduces this)

#if __has_builtin(__builtin_amdgcn_tensor_load_to_lds)
#define HAVE_TDM 1
#else
#define HAVE_TDM 0
#endif

// ---------------------------------------------------------------------------
// Prep: e2[g,m] = ||embeddings[g,m]||^2 into workspace; also initialize the
// count region of d_out from the input count (d_out is poisoned by harness).
// ---------------------------------------------------------------------------
__global__ __launch_bounds__(256) void vq_prep(const float* __restrict__ emb,
                                               const float* __restrict__ cnt_in,
                                               float* __restrict__ e2,
                                               float* __restrict__ cnt_out) {
  const int wave = threadIdx.x >> 5, lane = threadIdx.x & 31;
  const int row  = blockIdx.x * 8 + wave;                 // 0 .. G*M-1
  const float* p = emb + (size_t)row * D;
  float s = 0.f;
#pragma unroll
  for (int j = 0; j < D; j += 32) { float v = p[lane + j]; s += v * v; }
#pragma unroll
  for (int m = 16; m >= 1; m >>= 1) s += __shfl_xor(s, m, 32);
  if (lane == 0) { e2[row] = s; cnt_out[row] = cnt_in[row]; }
}

// ---------------------------------------------------------------------------
// Main kernel: per (g, 128-row n-tile) block, 8 waves. Each wave holds a
// 16-row A strip of x in registers (32 x v2f k-fragments). Embedding tiles
// (32 x 128) are DMA'd into double-buffered LDS by the Tensor Data Mover
// (TENSORcnt-pipelined), and each chunk is consumed by two interleaved
// V_WMMA_F32_16X16X4_F32 accumulator chains. Running per-lane argmin of
// e2 - 2*x.e; cross-lane reduce via LDS at the end.
// ---------------------------------------------------------------------------
struct Shm {
  float e_s[2][MTILE * EPAD];   // double-buffered embedding tiles (padded rows)
  float red_min[NTILE][16];
  int   red_idx[NTILE][16];
  int   idx_s[NTILE];
};

__global__ __launch_bounds__(256) void vq_main(const float* __restrict__ x,
                                               const float* __restrict__ emb,
                                               const float* __restrict__ e2w,
                                               float* __restrict__ xq,
                                               float* __restrict__ ind,
                                               float* __restrict__ cnt) {
  __shared__ Shm sh;

  const int tid   = threadIdx.x;
  const int wave  = tid >> 5, lane = tid & 31;
  const int g     = blockIdx.x & (G - 1);
  const int ntile = blockIdx.x >> 2;                       // G == 4
  const int n0    = ntile * NTILE;
  const int half  = lane >> 4, l16 = lane & 15;
  const bool issuer = (tid < 32);                          // wave 0 drives TDM

  const float* eg  = emb + (size_t)g * M * D;
  const float* e2g = e2w + (size_t)g * M;

  // --- preload A fragments: lane l holds x[row = wave*16 + l%16][k0..k0+1],
  // k0 = kk*4 + 2*(l/16)  (ISA 32-bit A 16x4 layout).
  v2f a[32];
  {
    const int    row_l = wave * 16 + l16;
    const size_t base  = ((size_t)(n0 + row_l) * TPD + g) * D;
#pragma unroll
    for (int kk = 0; kk < 32; ++kk) {
      const int k0 = kk * 4 + 2 * half;
      const float2 t = *reinterpret_cast<const float2*>(x + base + k0);
      a[kk] = (v2f){t.x, t.y};
    }
  }

  float runmin[8];
  int   runidx[8];
#pragma unroll
  for (int v = 0; v < 8; ++v) { runmin[v] = 3.4e38f; runidx[v] = 0; }

#if HAVE_TDM
  // Issue one TDM descriptor: 2-D tile (tile_dim0=128 elems, tile_dim1=32 rows),
  // 4-byte elements, row stride 128 elems, hardware LDS padding of 4 DWORDs
  // after every 128 DWORDs (=> 132-float padded rows in LDS).
  auto tdm_issue = [&](int mb, int buf) {
    const uint64_t ga = (uint64_t)(uintptr_t)(eg + (size_t)mb * D);
    const uint32_t lds_off =
        (uint32_t)((uintptr_t)&sh.e_s[buf][0] - (uintptr_t)&sh);
    u32x4 g0;
    g0[0] = 1u;                                   // count=1, user mode
    g0[1] = lds_off;                              // lds_addr
    g0[2] = (uint32_t)ga;                         // global_addr[31:0]
    g0[3] = (uint32_t)((ga >> 32) & 0x1FFFFFFull) // global_addr[56:32]
            | (2u << 30);                         // type = 2 ("image")
    i32x8 g1;
    g1[0] = (2 << 16)        // data_size = 4 bytes
          | (1 << 20)        // pad_enable
          | (6 << 22)        // pad_interval: 128 DWORDs
          | (3 << 25);       // pad_amount:   4 DWORDs
    g1[1] = (int)(128u << 16);   // tensor_dim0 = 128 (low16)
    g1[2] = (int)(4096u << 16);  // tensor_dim1 (low16)
    g1[3] = (int)(128u << 16);   // tile_dim0 = 128
    g1[4] = MTILE;               // tile_dim1 = 32, tile_dim2 = 0
    g1[5] = 128;                 // tensor_dim0_stride = 128 elems
    g1[6] = 0;
    g1[7] = 0;
    i32x4 z4 = {};
#if __clang_major__ >= 23
    i32x8 z8 = {};
    __builtin_amdgcn_tensor_load_to_lds(g0, g1, z4, z4, z8, 0);
#else
    __builtin_amdgcn_tensor_load_to_lds(g0, g1, z4, z4, 0);
#endif
  };
  if (issuer) tdm_issue(0, 0);
#endif

  for (int c = 0; c < M / MTILE; ++c) {
    const int mb  = c * MTILE;
    const int buf = c & 1;

#if HAVE_TDM
    if (issuer) __builtin_amdgcn_s_wait_tensorcnt(0);  // chunk c arrived
    __syncthreads();                                    // publish to all waves
    if (issuer && (mb + MTILE < M)) tdm_issue(mb + MTILE, buf ^ 1);
#else
    __syncthreads();
    {  // fallback: cooperative synchronous copy
      const float4* src = reinterpret_cast<const float4*>(eg + (size_t)mb * D);
#pragma unroll
      for (int i = 0; i < (MTILE * D / 4) / 256; ++i) {
        const int q = tid + i * 256;
        const int r = q >> 5, cc = q & 31;
        float4 v = src[q];
        *reinterpret_cast<float4*>(&sh.e_s[buf][r * EPAD + cc * 4]) = v;
      }
      if (mb + MTILE < M)
        __builtin_prefetch(eg + (size_t)(mb + MTILE) * D + tid * 32, 0, 1);
    }
    __syncthreads();
#endif

    // --- two interleaved WMMA accumulator chains over K = 128 (32 x K=4) ---
    v8f c0 = {}, c1 = {};
    const float* b0 = &sh.e_s[buf][(0  + l16) * EPAD + 2 * half];
    const float* b1 = &sh.e_s[buf][(16 + l16) * EPAD + 2 * half];
#pragma unroll
    for (int kk = 0; kk < 32; ++kk) {
      const float2 t0 = *reinterpret_cast<const float2*>(b0 + kk * 4);
      const float2 t1 = *reinterpret_cast<const float2*>(b1 + kk * 4);
      c0 = __builtin_amdgcn_wmma_f32_16x16x4_f32(false, a[kk], false,
                                                 (v2f){t0.x, t0.y}, (short)0,
                                                 c0, false, false);
      c1 = __builtin_amdgcn_wmma_f32_16x16x4_f32(false, a[kk], false,
                                                 (v2f){t1.x, t1.y}, (short)0,
                                                 c1, false, false);
    }
    const float e2v0 = e2g[mb + l16];
    const float e2v1 = e2g[mb + 16 + l16];
    const int   mg0  = mb + l16;
    const int   mg1  = mb + 16 + l16;
#pragma unroll
    for (int v = 0; v < 8; ++v) {
      const float s0 = fmaf(-2.0f, c0[v], e2v0);
      if (s0 < runmin[v]) { runmin[v] = s0; runidx[v] = mg0; }
      const float s1 = fmaf(-2.0f, c1[v], e2v1);
      if (s1 < runmin[v]) { runmin[v] = s1; runidx[v] = mg1; }
    }
  }

  // --- cross-lane argmin reduction via LDS ---
  __syncthreads();
#pragma unroll
  for (int v = 0; v < 8; ++v) {
    const int row_l = wave * 16 + v + 8 * half;  // C layout: row = v + 8*(l>=16)
    sh.red_min[row_l][l16] = runmin[v];
    sh.red_idx[row_l][l16] = runidx[v];
  }
  __syncthreads();
  if (tid < NTILE) {
    float best = sh.red_min[tid][0];
    int   bi   = sh.red_idx[tid][0];
#pragma unroll
    for (int j = 1; j < 16; ++j) {
      const float mv = sh.red_min[tid][j];
      const int   mi = sh.red_idx[tid][j];
      if (mv < best || (mv == best && mi < bi)) { best = mv; bi = mi; }
    }
    sh.idx_s[tid] = bi;
    const int n_g = n0 + tid;
    ind[(size_t)n_g * TPD + g] = (float)bi;
    atomicAdd(&cnt[(size_t)g * M + bi], 1.0f);
  }
  __syncthreads();

  // --- cooperative gather of winning embeddings into x_quant ---
  for (int i = tid; i < NTILE * (D / 4); i += 256) {
    const int r  = i >> 5;                 // 32 float4 per row
    const int c4 = i & 31;
    const int bi = sh.idx_s[r];
    const float4 v =
        reinterpret_cast<const float4*>(eg + (size_t)bi * D)[c4];
    reinterpret_cast<float4*>(xq + ((size_t)(n0 + r) * TPD + g) * D)[c4] = v;
  }
}

extern "C" void kernel_launch(void* const* d_in, const int* in_sizes, int n_in,
                              void* d_out, int out_size, void* d_ws, size_t ws_size,
                              hipStream_t stream) {
  const float* x      = (const float*)d_in[0];   // (16384, 4, 128)
  const float* emb    = (const float*)d_in[1];   // (4, 4096, 128)
  const float* cnt_in = (const float*)d_in[2];   // (4, 4096)

  float* out = (float*)d_out;
  float* xq  = out;                                // 16384*4*128 floats
  float* ind = out + (size_t)BS * TPD * D;         // 16384*4 (indices as float)
  float* cnt = ind + (size_t)BS * TPD;             // 4*4096
  float* e2  = (float*)d_ws;                       // 16384 floats scratch

  vq_prep<<<dim3((G * M) / 8), dim3(256), 0, stream>>>(emb, cnt_in, e2, cnt);
  vq_main<<<dim3(G * (BS / NTILE)), dim3(256), 0, stream>>>(x, emb, e2, xq, ind, cnt);
}